// ShapeNetModelSeg_19215683682642
// MI455X (gfx1250) — compile-verified
//
#include <hip/hip_runtime.h>

// ---------------- problem constants (from reference) ----------------
#define Bb   16
#define Nn   2048
#define Ss   512
#define KNNc 32
#define UPKc 3
#define C1c  128
#define C2c  256
#define CAc  128
#define Hh   8
#define FFc  512
#define DHd  16

typedef float v2f __attribute__((ext_vector_type(2)));
typedef float v8f __attribute__((ext_vector_type(8)));

__device__ __forceinline__ float leakyf(float x) { return x >= 0.f ? x : 0.2f * x; }

// order-preserving float->uint map (for packed-key argmin/argmax with index tiebreak)
__device__ __forceinline__ unsigned ord32(float f) {
  unsigned u = __float_as_uint(f);
  return (u & 0x80000000u) ? ~u : (u | 0x80000000u);
}
__device__ __forceinline__ unsigned long long umin64(unsigned long long a, unsigned long long b) { return a < b ? a : b; }
__device__ __forceinline__ unsigned long long umax64(unsigned long long a, unsigned long long b) { return a > b ? a : b; }

__device__ __forceinline__ v8f wmma4(v2f a, v2f b, v8f c) {
  return __builtin_amdgcn_wmma_f32_16x16x4_f32(false, a, false, b, (short)0, c, false, false);
}

// =====================================================================
// Generic strided, batched fp32 GEMM on the CDNA5 matrix pipe.
// C[m,n] (+)= sum_k A[m,k]*B[k,n].  A must be K-contiguous (all our
// operands are).  Per-lane source coordinates are loop-invariant, so they
// are clamped once up front (garbage from clamped lanes only reaches
// output elements discarded by the masked store) -> branch-free inner
// loop, EXEC all-ones around WMMA, pointer-bump addressing, and a 2-stage
// software pipeline so global_load_b64s overlap the matrix pipe.
// Fragment layouts per CDNA5 ISA 7.12.2:
//   A 16x4 : lane L -> m=L&15, kpair=2*(L>>4); vgpr j -> k=kpair+j
//   B 4x16 : lane L -> n=L&15, kpair=2*(L>>4); vgpr j -> k=kpair+j
//   C 16x16: vgpr r, lane L -> n=L&15, m=r+8*(L>>4)
// =====================================================================
template<bool BK1>
__device__ __forceinline__ void gemm_kloop(const float* pa0, const float* pa1,
                                           const float* pb0, const float* pb1,
                                           long sBk, int K, v8f (&acc)[2][2])
{
  const long stepB = 4 * sBk;
  v2f a0, a1, b0, b1;
  a0 = *(const v2f*)pa0;
  a1 = *(const v2f*)pa1;
  if (BK1) { b0 = *(const v2f*)pb0; b1 = *(const v2f*)pb1; }
  else     { b0.x = pb0[0]; b0.y = pb0[sBk]; b1.x = pb1[0]; b1.y = pb1[sBk]; }
  int steps = K >> 2;
  for (int it = 0; it < steps - 1; ++it) {
    pa0 += 4; pa1 += 4; pb0 += stepB; pb1 += stepB;
    v2f na0 = *(const v2f*)pa0;
    v2f na1 = *(const v2f*)pa1;
    v2f nb0, nb1;
    if (BK1) { nb0 = *(const v2f*)pb0; nb1 = *(const v2f*)pb1; }
    else     { nb0.x = pb0[0]; nb0.y = pb0[sBk]; nb1.x = pb1[0]; nb1.y = pb1[sBk]; }
    acc[0][0] = wmma4(a0, b0, acc[0][0]);
    acc[0][1] = wmma4(a0, b1, acc[0][1]);
    acc[1][0] = wmma4(a1, b0, acc[1][0]);
    acc[1][1] = wmma4(a1, b1, acc[1][1]);
    a0 = na0; a1 = na1; b0 = nb0; b1 = nb1;
  }
  acc[0][0] = wmma4(a0, b0, acc[0][0]);
  acc[0][1] = wmma4(a0, b1, acc[0][1]);
  acc[1][0] = wmma4(a1, b0, acc[1][0]);
  acc[1][1] = wmma4(a1, b1, acc[1][1]);
}

__global__ __launch_bounds__(128) void gemm_wmma_f32(
    const float* __restrict__ A, long sAm, long sAz,           // A[m,k] = A[m*sAm + k]
    const float* __restrict__ B, long sBk, long sBn, long sBz, // B[k,n] = B[k*sBk + n*sBn]
    float* C, long sCm, long sCn, long sCz,
    int M, int N, int K, int accumulate)
{
  long z = blockIdx.z;
  A += z * sAz; B += z * sBz; C += z * sCz;
  int wave = threadIdx.x >> 5;
  int lane = threadIdx.x & 31;
  int lm   = lane & 15;
  int kb   = (lane >> 4) << 1;
  int m0 = blockIdx.y * 64 + (wave >> 1) * 32;
  int n0 = blockIdx.x * 64 + (wave & 1) * 32;

  // loop-invariant, clamped per-lane source coordinates
  int ma0 = min(m0 + lm,      M - 1);
  int ma1 = min(m0 + 16 + lm, M - 1);
  int nb0 = min(n0 + lm,      N - 1);
  int nb1 = min(n0 + 16 + lm, N - 1);
  const float* pa0 = A + (long)ma0 * sAm + kb;
  const float* pa1 = A + (long)ma1 * sAm + kb;
  const float* pb0 = B + (long)kb * sBk + (long)nb0 * sBn;
  const float* pb1 = B + (long)kb * sBk + (long)nb1 * sBn;

  v8f acc[2][2] = {};
  if (sBk == 1) gemm_kloop<true >(pa0, pa1, pb0, pb1, sBk, K, acc);
  else          gemm_kloop<false>(pa0, pa1, pb0, pb1, sBk, K, acc);

  bool full = ((M & 63) == 0) && ((N & 63) == 0);   // uniform: no masking needed
  int mh = (lane >> 4) << 3;
#pragma unroll
  for (int i = 0; i < 2; ++i)
#pragma unroll
    for (int j = 0; j < 2; ++j)
#pragma unroll
      for (int r = 0; r < 8; ++r) {
        int m = m0 + i * 16 + mh + r;
        int n = n0 + j * 16 + lm;
        if (full || (m < M && n < N)) {
          long o = (long)m * sCm + (long)n * sCn;
          float v = acc[i][j][r];
          C[o] = accumulate ? C[o] + v : v;
        }
      }
}

// ---------------- tiny K=3 embed GEMM (not worth WMMA) ----------------
__global__ void emb1_kernel(const float* __restrict__ pcd, const float* __restrict__ w,
                            float* __restrict__ y, long total)
{
  long i = (long)blockIdx.x * blockDim.x + threadIdx.x;
  if (i >= total) return;
  int n = (int)(i % Nn); long t = i / Nn; int o = (int)(t % C1c); int b = (int)(t / C1c);
  float s = 0.f;
#pragma unroll
  for (int c = 0; c < 3; ++c) s += w[o * 3 + c] * pcd[((long)b * 3 + c) * Nn + n];
  y[i] = s;
}

// ---------------- batchnorm helpers (deterministic tree reductions) ----------------
// layout (B, C, N): one block per channel
__global__ __launch_bounds__(256) void bn_stats_cmajor(const float* __restrict__ y,
                                                       float* __restrict__ stats, int C, long Nlen)
{
  int c = blockIdx.x;
  __shared__ float s1[256], s2[256];
  long cnt = (long)Bb * Nlen;
  float a = 0.f, b = 0.f;
  for (long t = threadIdx.x; t < cnt; t += 256) {
    long bi = t / Nlen, n = t % Nlen;
    float v = y[(bi * C + c) * Nlen + n];
    a += v; b += v * v;
  }
  s1[threadIdx.x] = a; s2[threadIdx.x] = b; __syncthreads();
  for (int s = 128; s > 0; s >>= 1) {
    if (threadIdx.x < s) { s1[threadIdx.x] += s1[threadIdx.x + s]; s2[threadIdx.x] += s2[threadIdx.x + s]; }
    __syncthreads();
  }
  if (threadIdx.x == 0) {
    float mean = s1[0] / (float)cnt;
    float var  = s2[0] / (float)cnt - mean * mean;
    stats[c * 2] = mean; stats[c * 2 + 1] = rsqrtf(var + 1e-5f);
  }
}

__global__ void bn_apply_cmajor(const float* __restrict__ y, float* __restrict__ out,
                                const float* __restrict__ stats, int C, long Nlen, long total, int do_leaky)
{
  long i = (long)blockIdx.x * blockDim.x + threadIdx.x;
  if (i >= total) return;
  int c = (int)((i / Nlen) % C);
  float v = (y[i] - stats[c * 2]) * stats[c * 2 + 1];
  out[i] = do_leaky ? leakyf(v) : v;
}

// layout (rows, C) channel-last: one block per channel
__global__ __launch_bounds__(256) void bn_stats_clast(const float* __restrict__ y,
                                                      float* __restrict__ stats, int C, long rows)
{
  int c = blockIdx.x;
  __shared__ float s1[256], s2[256];
  float a = 0.f, b = 0.f;
  for (long r = threadIdx.x; r < rows; r += 256) { float v = y[r * C + c]; a += v; b += v * v; }
  s1[threadIdx.x] = a; s2[threadIdx.x] = b; __syncthreads();
  for (int s = 128; s > 0; s >>= 1) {
    if (threadIdx.x < s) { s1[threadIdx.x] += s1[threadIdx.x + s]; s2[threadIdx.x] += s2[threadIdx.x + s]; }
    __syncthreads();
  }
  if (threadIdx.x == 0) {
    float mean = s1[0] / (float)rows;
    float var  = s2[0] / (float)rows - mean * mean;
    stats[c * 2] = mean; stats[c * 2 + 1] = rsqrtf(var + 1e-5f);
  }
}

__global__ void bn_apply_clast(const float* __restrict__ y, float* __restrict__ out,
                               const float* __restrict__ stats, int C, long total, int do_leaky)
{
  long i = (long)blockIdx.x * blockDim.x + threadIdx.x;
  if (i >= total) return;
  int c = (int)(i % C);
  float v = (y[i] - stats[c * 2]) * stats[c * 2 + 1];
  out[i] = do_leaky ? leakyf(v) : v;
}

// ---------------- misc elementwise ----------------
__global__ void add_kernel(const float* __restrict__ a, const float* __restrict__ b,
                           float* __restrict__ o, long total)
{
  long i = (long)blockIdx.x * blockDim.x + threadIdx.x;
  if (i < total) o[i] = a[i] + b[i];
}
__global__ void leaky_kernel(float* __restrict__ x, long total)
{
  long i = (long)blockIdx.x * blockDim.x + threadIdx.x;
  if (i < total) x[i] = leakyf(x[i]);
}
__global__ void wab_kernel(const float* __restrict__ w_ec1, float* __restrict__ wab, long total)
{ // wab[o][c] = w_ec1[o][c] - w_ec1[o][256+c]
  long i = (long)blockIdx.x * blockDim.x + threadIdx.x;
  if (i >= total) return;
  int c = (int)(i % C2c); int o = (int)(i / C2c);
  wab[i] = w_ec1[(long)o * 512 + c] - w_ec1[(long)o * 512 + C2c + c];
}
__global__ void xnorm_kernel(const float* __restrict__ x2, float* __restrict__ xn, long total)
{ // xn[b][n] = sum_c x2[b][c][n]^2
  long i = (long)blockIdx.x * blockDim.x + threadIdx.x;
  if (i >= total) return;
  int b = (int)(i / Nn), n = (int)(i % Nn);
  const float* p = x2 + (long)b * C2c * Nn + n;
  float s = 0.f;
  for (int c = 0; c < C2c; ++c) { float v = p[(long)c * Nn]; s += v * v; }
  xn[i] = s;
}

// ---------------- farthest point sampling (serial critical path) ----------------
// one block per batch; 1024 threads, 2 points each; dmin lives in registers.
__global__ __launch_bounds__(1024) void fps_kernel(const float* __restrict__ x, int* __restrict__ idx)
{
  int b = blockIdx.x;
  const float* xb = x + (long)b * C2c * Nn;
  __shared__ unsigned long long red[1024];
  __shared__ int s_last;
  int tid = threadIdx.x;
  float dmin0 = __builtin_inff(), dmin1 = __builtin_inff();
  if (tid == 0) s_last = 0;
  __syncthreads();
  for (int s = 0; s < Ss; ++s) {
    int last = s_last;
    if (tid == 0) idx[b * Ss + s] = last;
    unsigned long long best = 0ull;
#pragma unroll
    for (int rep = 0; rep < 2; ++rep) {
      int n = tid + rep * 1024;
      float d = 0.f;
      for (int c = 0; c < C2c; ++c) {
        float diff = xb[(long)c * Nn + n] - xb[(long)c * Nn + last];
        d += diff * diff;
      }
      float dm = fminf(rep == 0 ? dmin0 : dmin1, d);
      if (rep == 0) dmin0 = dm; else dmin1 = dm;
      // argmax of dmin; ties -> smallest index (jnp.argmax semantics)
      unsigned long long key = ((unsigned long long)ord32(dm) << 32)
                             | (unsigned long long)(0xFFFFFFFFu - (unsigned)n);
      best = umax64(best, key);
    }
    red[tid] = best; __syncthreads();
    for (int off = 512; off > 0; off >>= 1) {
      if (tid < off) red[tid] = umax64(red[tid], red[tid + off]);
      __syncthreads();
    }
    if (tid == 0) s_last = (int)(0xFFFFFFFFu - (unsigned)(red[0] & 0xFFFFFFFFull));
    __syncthreads();
  }
}

// ---------------- gathers ----------------
__global__ void gatherq_kernel(const float* __restrict__ x2, const float* __restrict__ xn,
                               const int* __restrict__ idx, float* __restrict__ qfeat,
                               float* __restrict__ qn, long total)
{
  long i = (long)blockIdx.x * blockDim.x + threadIdx.x;
  if (i >= total) return;
  int c = (int)(i % C2c); long bs = i / C2c; int b = (int)(bs / Ss);
  int id = idx[bs];
  qfeat[i] = x2[((long)b * C2c + c) * Nn + id];
  if (c == 0) qn[bs] = xn[(long)b * Nn + id];
}
__global__ void gatherxyz_kernel(const float* __restrict__ pcd, const int* __restrict__ idx,
                                 float* __restrict__ sxyz, long total)
{
  long i = (long)blockIdx.x * blockDim.x + threadIdx.x;
  if (i >= total) return;
  int c = (int)(i % 3); long bs = i / 3; int b = (int)(bs / Ss);
  sxyz[i] = pcd[((long)b * 3 + c) * Nn + idx[bs]];
}

// ---------------- top-32 neighbor selection ----------------
// one block per (b,s); iterative argmin with (value,index) packed keys — exact
// lax.top_k tie-breaking (smaller index wins on equal distance).
__global__ __launch_bounds__(256) void topk32_kernel(const float* __restrict__ dot,
    const float* __restrict__ xn, const float* __restrict__ qn, int* __restrict__ nidx)
{
  long bs = blockIdx.x;
  int b = (int)(bs / Ss);
  __shared__ float d2s[Nn];
  __shared__ unsigned long long rk[256];
  float q = qn[bs];
  for (int n = threadIdx.x; n < Nn; n += 256)
    d2s[n] = q - 2.f * dot[bs * Nn + n] + xn[(long)b * Nn + n];
  __syncthreads();
  for (int kk = 0; kk < KNNc; ++kk) {
    unsigned long long best = ~0ull;
    for (int n = threadIdx.x; n < Nn; n += 256) {
      unsigned long long key = ((unsigned long long)ord32(d2s[n]) << 32) | (unsigned)n;
      best = umin64(best, key);
    }
    rk[threadIdx.x] = best; __syncthreads();
    for (int s = 128; s > 0; s >>= 1) {
      if (threadIdx.x < s) rk[threadIdx.x] = umin64(rk[threadIdx.x], rk[threadIdx.x + s]);
      __syncthreads();
    }
    if (threadIdx.x == 0) {
      int w = (int)(rk[0] & 0xFFFFFFFFull);
      nidx[bs * KNNc + kk] = w;
      d2s[w] = __builtin_inff();
    }
    __syncthreads();
  }
}

// ---------------- fused edge-conv gather + k-max + BN partial stats ----------------
// h1[b,s,k,o] = t1[b,s,o] + t2t[b, nidx[b,s,k], o]; never materialized.
__global__ __launch_bounds__(128) void edge_kernel(const float* __restrict__ t1,
    const float* __restrict__ t2t, const int* __restrict__ nidx,
    float* __restrict__ hmax, float* __restrict__ partial)
{
  long bs = blockIdx.x;
  int b = (int)(bs / Ss);
  int o = threadIdx.x;
  float base = t1[bs * CAc + o];
  const int* ni = nidx + bs * KNNc;
  float mx = -__builtin_inff(), s = 0.f, s2 = 0.f;
  for (int k = 0; k < KNNc; ++k) {
    float v = base + t2t[((long)b * Nn + ni[k]) * CAc + o];
    mx = fmaxf(mx, v); s += v; s2 += v * v;
  }
  hmax[bs * CAc + o] = mx;
  partial[(bs * CAc + o) * 2 + 0] = s;
  partial[(bs * CAc + o) * 2 + 1] = s2;
}

__global__ __launch_bounds__(256) void edge_stats_kernel(const float* __restrict__ partial,
                                                         float* __restrict__ stats)
{
  int o = blockIdx.x;
  __shared__ float s1[256], s2[256];
  float a = 0.f, b = 0.f;
  long nblk = (long)Bb * Ss;
  for (long r = threadIdx.x; r < nblk; r += 256) {
    a += partial[(r * CAc + o) * 2 + 0];
    b += partial[(r * CAc + o) * 2 + 1];
  }
  s1[threadIdx.x] = a; s2[threadIdx.x] = b; __syncthreads();
  for (int s = 128; s > 0; s >>= 1) {
    if (threadIdx.x < s) { s1[threadIdx.x] += s1[threadIdx.x + s]; s2[threadIdx.x] += s2[threadIdx.x + s]; }
    __syncthreads();
  }
  if (threadIdx.x == 0) {
    float cnt = (float)((long)Bb * Ss * KNNc);
    float mean = s1[0] / cnt;
    float var  = s2[0] / cnt - mean * mean;
    stats[o * 2] = mean; stats[o * 2 + 1] = rsqrtf(var + 1e-5f);
  }
}

// ---------------- attention helpers ----------------
__global__ void t2t_kernel(const float* __restrict__ t2, float* __restrict__ t2t, long total)
{ // (B,C,N) -> (B,N,C)
  long i = (long)blockIdx.x * blockDim.x + threadIdx.x;
  if (i >= total) return;
  int c = (int)(i % CAc); long bn = i / CAc; int b = (int)(bn / Nn), n = (int)(bn % Nn);
  t2t[i] = t2[((long)b * CAc + c) * Nn + n];
}
__global__ void split_heads_kernel(const float* __restrict__ in, float* __restrict__ out, long total)
{ // (B,S,CA) -> (B,H,S,DH)
  long i = (long)blockIdx.x * blockDim.x + threadIdx.x;
  if (i >= total) return;
  int d = (int)(i % DHd); long t = i / DHd; int s = (int)(t % Ss); t /= Ss;
  int h = (int)(t % Hh); int b = (int)(t / Hh);
  out[i] = in[((long)b * Ss + s) * CAc + h * DHd + d];
}
__global__ void merge_heads_kernel(const float* __restrict__ in, float* __restrict__ out, long total)
{ // (B,H,S,DH) -> (B,S,CA)
  long i = (long)blockIdx.x * blockDim.x + threadIdx.x;
  if (i >= total) return;
  int c = (int)(i % CAc); long t = i / CAc; int s = (int)(t % Ss); int b = (int)(t / Ss);
  int h = c / DHd, d = c % DHd;
  out[i] = in[(((long)b * Hh + h) * Ss + s) * DHd + d];
}
__global__ __launch_bounds__(256) void softmax_kernel(float* __restrict__ att, float scale)
{
  long row = blockIdx.x;
  float* p = att + row * (long)Ss;
  __shared__ float red[256];
  int tid = threadIdx.x;
  float m = -__builtin_inff();
  for (int n = tid; n < Ss; n += 256) m = fmaxf(m, p[n] * scale);
  red[tid] = m; __syncthreads();
  for (int off = 128; off > 0; off >>= 1) { if (tid < off) red[tid] = fmaxf(red[tid], red[tid + off]); __syncthreads(); }
  float mx = red[0]; __syncthreads();
  float s = 0.f;
  for (int n = tid; n < Ss; n += 256) { float e = expf(p[n] * scale - mx); p[n] = e; s += e; }
  red[tid] = s; __syncthreads();
  for (int off = 128; off > 0; off >>= 1) { if (tid < off) red[tid] += red[tid + off]; __syncthreads(); }
  float inv = 1.f / red[0]; __syncthreads();
  for (int n = tid; n < Ss; n += 256) p[n] *= inv;
}

// ---------------- 3-NN upsampling ----------------
__global__ void knn3_kernel(const float* __restrict__ pcd, const float* __restrict__ sxyz,
                            int* __restrict__ nidx3, float* __restrict__ w3, long total)
{
  long i = (long)blockIdx.x * blockDim.x + threadIdx.x;
  if (i >= total) return;
  int b = (int)(i / Nn), n = (int)(i % Nn);
  float px = pcd[((long)b * 3 + 0) * Nn + n];
  float py = pcd[((long)b * 3 + 1) * Nn + n];
  float pz = pcd[((long)b * 3 + 2) * Nn + n];
  const float* sx = sxyz + (long)b * Ss * 3;
  float v0 = __builtin_inff(), v1 = v0, v2 = v0;
  int i0 = 0, i1 = 0, i2 = 0;
  for (int s = 0; s < Ss; ++s) {
    float dx = px - sx[s * 3 + 0], dy = py - sx[s * 3 + 1], dz = pz - sx[s * 3 + 2];
    float d2 = dx * dx + dy * dy + dz * dz;
    if (d2 < v0)      { v2 = v1; i2 = i1; v1 = v0; i1 = i0; v0 = d2; i0 = s; }
    else if (d2 < v1) { v2 = v1; i2 = i1; v1 = d2; i1 = s; }
    else if (d2 < v2) { v2 = d2; i2 = s; }
  }
  float w0 = 1.f / (sqrtf(fmaxf(v0, 1e-12f)) + 1e-8f);
  float w1 = 1.f / (sqrtf(fmaxf(v1, 1e-12f)) + 1e-8f);
  float w2 = 1.f / (sqrtf(fmaxf(v2, 1e-12f)) + 1e-8f);
  float inv = 1.f / (w0 + w1 + w2);
  nidx3[i * 3 + 0] = i0; nidx3[i * 3 + 1] = i1; nidx3[i * 3 + 2] = i2;
  w3[i * 3 + 0] = w0 * inv; w3[i * 3 + 1] = w1 * inv; w3[i * 3 + 2] = w2 * inv;
}

__global__ __launch_bounds__(128) void interp_kernel(const float* __restrict__ Z2,
    const int* __restrict__ nidx3, const float* __restrict__ w3, float* __restrict__ interp)
{
  long bn = blockIdx.x;              // b*Nn + n
  int b = (int)(bn / Nn);
  int c = threadIdx.x;
  const int* ni = nidx3 + bn * 3;
  const float* w = w3 + bn * 3;
  float v = 0.f;
#pragma unroll
  for (int j = 0; j < 3; ++j) v += w[j] * Z2[((long)b * Ss + ni[j]) * CAc + c];
  interp[bn * CAc + c] = v;
}

// =====================================================================
// host side
// =====================================================================
static void gemm(hipStream_t st,
                 const float* A, long sAm, long sAz,
                 const float* B, long sBk, long sBn, long sBz,
                 float* C, long sCm, long sCn, long sCz,
                 int M, int N, int K, int Z, int acc)
{
  dim3 g((N + 63) / 64, (M + 63) / 64, Z);
  gemm_wmma_f32<<<g, 128, 0, st>>>(A, sAm, sAz, B, sBk, sBn, sBz, C, sCm, sCn, sCz, M, N, K, acc);
}
static inline long cdiv(long a, long b) { return (a + b - 1) / b; }

extern "C" void kernel_launch(void* const* d_in, const int* in_sizes, int n_in,
                              void* d_out, int out_size, void* d_ws, size_t ws_size,
                              hipStream_t stream)
{
  (void)in_sizes; (void)n_in; (void)out_size;
  const float* pcd    = (const float*)d_in[0];
  const float* w_emb1 = (const float*)d_in[1];
  const float* w_emb2 = (const float*)d_in[2];
  const float* w_ec1  = (const float*)d_in[3];
  const float* w_ec2  = (const float*)d_in[4];
  const float* w_q    = (const float*)d_in[5];
  const float* w_k    = (const float*)d_in[6];
  const float* w_v    = (const float*)d_in[7];
  const float* w_ff1  = (const float*)d_in[8];
  const float* w_ff2  = (const float*)d_in[9];
  const float* w_up   = (const float*)d_in[10];
  float* outp = (float*)d_out;

  char* base = (char*)d_ws;
  size_t off = 0;
  auto allocF = [&](size_t n) { float* p = (float*)(base + off); off += ((n * 4 + 255) / 256) * 256; return p; };
  auto allocI = [&](size_t n) { int*   p = (int*)  (base + off); off += ((n * 4 + 255) / 256) * 256; return p; };

  float* x1    = allocF((size_t)Bb * C1c * Nn);
  float* x2    = allocF((size_t)Bb * C2c * Nn);
  float* xn    = allocF((size_t)Bb * Nn);
  float* qfeat = allocF((size_t)Bb * Ss * C2c);
  float* qn    = allocF((size_t)Bb * Ss);
  float* Wab   = allocF((size_t)CAc * C2c);
  float* t1    = allocF((size_t)Bb * Ss * CAc);
  float* t2    = allocF((size_t)Bb * CAc * Nn);
  float* t2t   = allocF((size_t)Bb * Nn * CAc);
  float* hmax  = allocF((size_t)Bb * Ss * CAc);
  float* Fq    = allocF((size_t)Bb * Ss * CAc);
  float* Qm    = allocF((size_t)Bb * Ss * CAc);
  float* Km    = allocF((size_t)Bb * Ss * CAc);
  float* Vm    = allocF((size_t)Bb * Ss * CAc);
  float* Qh    = allocF((size_t)Bb * Ss * CAc);
  float* Kh    = allocF((size_t)Bb * Ss * CAc);
  float* Vh    = allocF((size_t)Bb * Ss * CAc);
  float* Oh    = allocF((size_t)Bb * Ss * CAc);
  float* Om    = allocF((size_t)Bb * Ss * CAc);
  float* Z1    = allocF((size_t)Bb * Ss * CAc);
  float* ffb   = allocF((size_t)Bb * Ss * FFc);
  float* ff2b  = allocF((size_t)Bb * Ss * CAc);
  float* Z2    = allocF((size_t)Bb * Ss * CAc);
  float* itp   = allocF((size_t)Bb * Nn * CAc);
  float* sxyz  = allocF((size_t)Bb * Ss * 3);
  float* w3    = allocF((size_t)Bb * Nn * 3);
  float* stats = allocF(2 * C2c);
  // arena shared by: dot (B*S*N), attention scores (B*H*S*S), edge partials (B*S*CA*2)
  size_t arenaN = (size_t)Bb * Hh * Ss * Ss;           // 33.6M floats (largest)
  float* arena = allocF(arenaN);
  int* idx   = allocI((size_t)Bb * Ss);
  int* nidx  = allocI((size_t)Bb * Ss * KNNc);
  int* nidx3 = allocI((size_t)Bb * Nn * UPKc);
  if (off > ws_size) return;                            // workspace too small: bail safely

  const long NB = Nn;
  long tot;

  // ---- stage 1: embed1 + BN + leaky ----
  tot = (long)Bb * C1c * Nn;
  emb1_kernel<<<cdiv(tot, 256), 256, 0, stream>>>(pcd, w_emb1, x1, tot);
  bn_stats_cmajor<<<C1c, 256, 0, stream>>>(x1, stats, C1c, NB);
  bn_apply_cmajor<<<cdiv(tot, 256), 256, 0, stream>>>(x1, x1, stats, C1c, NB, tot, 1);

  // ---- stage 2: embed2 GEMM (WMMA) + BN + leaky ----
  gemm(stream, w_emb2, C1c, 0, x1, NB, 1, (long)C1c * Nn,
       x2, NB, 1, (long)C2c * Nn, C2c, Nn, C1c, Bb, 0);
  tot = (long)Bb * C2c * Nn;
  bn_stats_cmajor<<<C2c, 256, 0, stream>>>(x2, stats, C2c, NB);
  bn_apply_cmajor<<<cdiv(tot, 256), 256, 0, stream>>>(x2, x2, stats, C2c, NB, tot, 1);

  // ---- stage 3: FPS on 256-d features ----
  tot = (long)Bb * Nn;
  xnorm_kernel<<<cdiv(tot, 256), 256, 0, stream>>>(x2, xn, tot);
  fps_kernel<<<Bb, 1024, 0, stream>>>(x2, idx);
  tot = (long)Bb * Ss * C2c;
  gatherq_kernel<<<cdiv(tot, 256), 256, 0, stream>>>(x2, xn, idx, qfeat, qn, tot);

  // ---- stage 4: KNN(32): distance GEMM (WMMA) + exact iterative top-k ----
  float* dotm = arena;
  gemm(stream, qfeat, C2c, (long)Ss * C2c, x2, NB, 1, (long)C2c * Nn,
       dotm, NB, 1, (long)Ss * Nn, Ss, Nn, C2c, Bb, 0);
  topk32_kernel<<<Bb * Ss, 256, 0, stream>>>(dotm, xn, qn, nidx);

  // ---- stage 5: edge conv, algebraically split: h1 = (Wa-Wb)q + Wb x[n] ----
  tot = (long)CAc * C2c;
  wab_kernel<<<cdiv(tot, 256), 256, 0, stream>>>(w_ec1, Wab, tot);
  gemm(stream, qfeat, C2c, (long)Ss * C2c, Wab, 1, C2c, 0,
       t1, CAc, 1, (long)Ss * CAc, Ss, CAc, C2c, Bb, 0);
  gemm(stream, w_ec1 + C2c, 2 * C2c, 0, x2, NB, 1, (long)C2c * Nn,
       t2, NB, 1, (long)CAc * Nn, CAc, Nn, C2c, Bb, 0);
  tot = (long)Bb * Nn * CAc;
  t2t_kernel<<<cdiv(tot, 256), 256, 0, stream>>>(t2, t2t, tot);
  float* partial = arena;                                // dot matrix is dead now
  edge_kernel<<<Bb * Ss, CAc, 0, stream>>>(t1, t2t, nidx, hmax, partial);
  edge_stats_kernel<<<CAc, 256, 0, stream>>>(partial, stats);
  tot = (long)Bb * Ss * CAc;
  bn_apply_clast<<<cdiv(tot, 256), 256, 0, stream>>>(hmax, hmax, stats, CAc, tot, 1);

  // ---- stage 6: ec2 GEMM + BN + leaky -> feat ----
  gemm(stream, hmax, CAc, (long)Ss * CAc, w_ec2, 1, CAc, 0,
       Fq, CAc, 1, (long)Ss * CAc, Ss, CAc, CAc, Bb, 0);
  bn_stats_clast<<<CAc, 256, 0, stream>>>(Fq, stats, CAc, (long)Bb * Ss);
  bn_apply_clast<<<cdiv(tot, 256), 256, 0, stream>>>(Fq, Fq, stats, CAc, tot, 1);

  // ---- stage 7: attention ----
  gemm(stream, Fq, CAc, (long)Ss * CAc, w_q, 1, CAc, 0,
       Qm, CAc, 1, (long)Ss * CAc, Ss, CAc, CAc, Bb, 0);
  gemm(stream, Fq, CAc, (long)Ss * CAc, w_k, 1, CAc, 0,
       Km, CAc, 1, (long)Ss * CAc, Ss, CAc, CAc, Bb, 0);
  gemm(stream, Fq, CAc, (long)Ss * CAc, w_v, 1, CAc, 0,
       Vm, CAc, 1, (long)Ss * CAc, Ss, CAc, CAc, Bb, 0);
  tot = (long)Bb * Hh * Ss * DHd;
  split_heads_kernel<<<cdiv(tot, 256), 256, 0, stream>>>(Qm, Qh, tot);
  split_heads_kernel<<<cdiv(tot, 256), 256, 0, stream>>>(Km, Kh, tot);
  split_heads_kernel<<<cdiv(tot, 256), 256, 0, stream>>>(Vm, Vh, tot);
  float* att = arena;                                    // edge partials are dead
  gemm(stream, Qh, DHd, (long)Ss * DHd, Kh, 1, DHd, (long)Ss * DHd,
       att, Ss, 1, (long)Ss * Ss, Ss, Ss, DHd, Bb * Hh, 0);
  softmax_kernel<<<Bb * Hh * Ss, 256, 0, stream>>>(att, 0.25f);
  gemm(stream, att, Ss, (long)Ss * Ss, Vh, DHd, 1, (long)Ss * DHd,
       Oh, DHd, 1, (long)Ss * DHd, Ss, DHd, Ss, Bb * Hh, 0);
  tot = (long)Bb * Ss * CAc;
  merge_heads_kernel<<<cdiv(tot, 256), 256, 0, stream>>>(Oh, Om, tot);

  // ---- stage 8: residual + BN; FF; residual + BN ----
  add_kernel<<<cdiv(tot, 256), 256, 0, stream>>>(Fq, Om, Z1, tot);
  bn_stats_clast<<<CAc, 256, 0, stream>>>(Z1, stats, CAc, (long)Bb * Ss);
  bn_apply_clast<<<cdiv(tot, 256), 256, 0, stream>>>(Z1, Z1, stats, CAc, tot, 0);
  gemm(stream, Z1, CAc, (long)Ss * CAc, w_ff1, 1, CAc, 0,
       ffb, FFc, 1, (long)Ss * FFc, Ss, FFc, CAc, Bb, 0);
  long totf = (long)Bb * Ss * FFc;
  leaky_kernel<<<cdiv(totf, 256), 256, 0, stream>>>(ffb, totf);
  gemm(stream, ffb, FFc, (long)Ss * FFc, w_ff2, 1, FFc, 0,
       ff2b, CAc, 1, (long)Ss * CAc, Ss, CAc, FFc, Bb, 0);
  add_kernel<<<cdiv(tot, 256), 256, 0, stream>>>(Z1, ff2b, Z2, tot);
  bn_stats_clast<<<CAc, 256, 0, stream>>>(Z2, stats, CAc, (long)Bb * Ss);
  bn_apply_clast<<<cdiv(tot, 256), 256, 0, stream>>>(Z2, Z2, stats, CAc, tot, 0);

  // ---- stage 9: 3-NN inverse-distance upsampling ----
  tot = (long)Bb * Ss * 3;
  gatherxyz_kernel<<<cdiv(tot, 256), 256, 0, stream>>>(pcd, idx, sxyz, tot);
  tot = (long)Bb * Nn;
  knn3_kernel<<<cdiv(tot, 256), 256, 0, stream>>>(pcd, sxyz, nidx3, w3, tot);
  interp_kernel<<<Bb * Nn, CAc, 0, stream>>>(Z2, nidx3, w3, itp);

  // ---- stage 10: final GEMM (split over concat) + BN + leaky -> d_out ----
  gemm(stream, w_up, C2c + CAc, 0, x2, NB, 1, (long)C2c * Nn,
       outp, NB, 1, (long)CAc * Nn, CAc, Nn, C2c, Bb, 0);
  gemm(stream, w_up + C2c, C2c + CAc, 0, itp, 1, CAc, (long)Nn * CAc,
       outp, NB, 1, (long)CAc * Nn, CAc, Nn, CAc, Bb, 1);
  tot = (long)Bb * CAc * Nn;
  bn_stats_cmajor<<<CAc, 256, 0, stream>>>(outp, stats, CAc, NB);
  bn_apply_cmajor<<<cdiv(tot, 256), 256, 0, stream>>>(outp, outp, stats, CAc, NB, tot, 1);
}